// Model_50568944943726
// MI455X (gfx1250) — compile-verified
//
#include <hip/hip_runtime.h>
#include <math.h>

#define B_  8
#define C_  96
#define J_  17
#define K_  96
#define NG_ 64      // PY*PX
#define W_  128

typedef __attribute__((ext_vector_type(2))) float v2f;
typedef __attribute__((ext_vector_type(8))) float v8f;

// ---------------------------------------------------------------------------
// Pass 0: per (b,k) pack WMMA B-fragments and visibility sums.
//   kp4[b][k][j] = ( -2*kx, -2*ky, kx^2+ky^2, vis ? 1/denom : 0 )
//   visSum[b][k] = max(sum_j vis, 1)
// ---------------------------------------------------------------------------
__global__ __launch_bounds__(256) void oks_prep(
    const float* __restrict__ keypoints,   // [B,K,J,3]
    const float* __restrict__ areas,       // [B,K]
    const float* __restrict__ sigmas,      // [J]
    float4* __restrict__ kp4,              // [B,K,J]
    float*  __restrict__ visSum)           // [B,K]
{
    int idx = blockIdx.x * blockDim.x + threadIdx.x;
    if (idx >= B_ * K_) return;
    float area = areas[idx];
    float vsum = 0.f;
    for (int j = 0; j < J_; ++j) {
        const float* kp = keypoints + (size_t)(idx * J_ + j) * 3;
        float kx = kp[0], ky = kp[1];
        float vis = (kp[2] > 0.f) ? 1.f : 0.f;
        float s2 = sigmas[j] * 2.f;
        float denom = fmaxf(2.f * s2 * s2 * area, 1e-6f);
        float4 o;
        o.x = -2.f * kx;
        o.y = -2.f * ky;
        o.z = kx * kx + ky * ky;
        o.w = (vis > 0.f) ? (1.f / denom) : 0.f;   // invDenom doubles as vis flag
        kp4[idx * J_ + j] = o;
        vsum += vis;
    }
    visSum[idx] = fmaxf(vsum, 1.f);
}

// ---------------------------------------------------------------------------
// Main: one 256-thread block (8 wave32) per (b,c).
//   Stage 1: transform 64*17 pose points; stage pre-packed A fragments in LDS:
//            sA0[j][n] = (x, y)       -> lanes 0-15  (K=0/1)
//            sA1[j][n] = (x^2+y^2, 1) -> lanes 16-31 (K=2/3)
//   Stage 2: WMMA f32 16x16x4 pairwise distances, grid-min, exp, accumulate.
//   Stage 3: argmax over K -> assignment.
//   Stage 4: OKS maps for assigned GT -> oks_parts_sel / oks_person_sel.
//   Stage 5: pose_gt.
// ---------------------------------------------------------------------------
__global__ __launch_bounds__(256) void oks_main(
    const float* __restrict__ pose_pool,      // [B,C,PY,PX,J,2]
    const float* __restrict__ keypoints,      // [B,K,J,3]
    const float* __restrict__ areas,          // [B,K]
    const float* __restrict__ transforms,     // [B,3,3]
    const float* __restrict__ transforms_inv, // [B,3,3]
    const int*   __restrict__ hflip,          // [B]
    const float* __restrict__ sigmas,         // [J]
    const float4* __restrict__ kp4,           // [B,K,J]
    const float* __restrict__ visSum,         // [B,K]
    float* __restrict__ out_parts,            // [B,C,PY,PX,J]
    float* __restrict__ out_person,           // [B,C,PY,PX]
    float* __restrict__ out_pose)             // [B,C,J,3]
{
    __shared__ v2f   sA0[J_ * NG_];   // (x, y)
    __shared__ v2f   sA1[J_ * NG_];   // (x^2+y^2, 1)
    __shared__ float acc[K_];
    __shared__ float sPerson[NG_];
    __shared__ int   sSel;

    const int bc  = blockIdx.x;
    const int b   = bc / C_;
    const int tid = threadIdx.x;

    if (tid < K_)  acc[tid]     = 0.f;
    if (tid < NG_) sPerson[tid] = 0.f;

    const float* ti = transforms_inv + b * 9;
    const float ti00 = ti[0], ti01 = ti[1], ti02 = ti[2];
    const float ti10 = ti[3], ti11 = ti[4], ti12 = ti[5];
    const bool flip = hflip[b] > 0;

    // ---- Stage 1: pose grid -> image coordinates, packed A fragments ----
    const float* pp = pose_pool + (size_t)bc * NG_ * J_ * 2;
    for (int i = tid; i < NG_ * J_; i += 256) {
        int n = i / J_;
        int j = i - n * J_;
        float px = pp[2 * i + 0];
        float py = pp[2 * i + 1];
        if (flip) px = (float)(W_ - 1) - px;
        float x = ti00 * px + ti01 * py + ti02;
        float y = ti10 * px + ti11 * py + ti12;
        int idx = j * NG_ + n;
        v2f lo; lo.x = x;         lo.y = y;
        v2f hi; hi.x = x*x + y*y; hi.y = 1.0f;
        sA0[idx] = lo;
        sA1[idx] = hi;
    }
    __syncthreads();

    // ---- Stage 2: WMMA distance tiles, grid-min, exp, accumulate over j ----
    const int wave   = tid >> 5;
    const int lane   = tid & 31;
    const int lane16 = lane & 15;
    const bool loHalf = (lane < 16);
    // branchless per-lane A-fragment base: address select, not data select
    const v2f* sAbase = loHalf ? sA0 : sA1;
    const float4* kpB = kp4 + (size_t)b * K_ * J_;

    for (int t = wave; t < J_ * 6; t += 8) {        // wave-uniform task loop
        int j  = t / 6;
        int kt = t - j * 6;
        int k16 = kt * 16 + lane16;

        float4 kv = kpB[(size_t)k16 * J_ + j];
        // B fragment (4x16, KxN): lanes 0-15 hold K=0/1 rows, 16-31 hold K=2/3.
        v2f bfrag;
        bfrag.x = loHalf ? kv.x : 1.0f;
        bfrag.y = loHalf ? kv.y : kv.z;

        const v2f* sArow = sAbase + j * NG_ + lane16;
        float md = 3.4e38f;
        #pragma unroll
        for (int m = 0; m < 4; ++m) {
            v2f afrag = sArow[m << 4];              // one ds_load_b64, no branch

            v8f cz = {0.f, 0.f, 0.f, 0.f, 0.f, 0.f, 0.f, 0.f};
            v8f d = __builtin_amdgcn_wmma_f32_16x16x4_f32(
                        false, afrag, false, bfrag, (short)0, cz, false, false);
            float m0 = fminf(fminf(d[0], d[1]), fminf(d[2], d[3]));
            float m1 = fminf(fminf(d[4], d[5]), fminf(d[6], d[7]));
            md = fminf(md, fminf(m0, m1));
        }
        // combine the two M-halves (lanes L and L+16 share column N=L%16)
        md = fminf(md, __shfl_xor(md, 16, 32));

        if (lane < 16) {
            float best = (kv.w > 0.f) ? expf(-md * kv.w) : 0.f; // max_grid oks
            atomicAdd(&acc[k16], best);
        }
    }
    __syncthreads();

    // ---- Stage 3: argmax over K (assignment) ----
    if (tid == 0) {
        float bestv = -1.f;
        int bi = 0;
        for (int k = 0; k < K_; ++k) {
            float v = acc[k] / visSum[b * K_ + k];
            if (v > bestv) { bestv = v; bi = k; }
        }
        sSel = bi;
    }
    __syncthreads();

    const int ksel = sSel;
    const float4* kpS = kpB + (size_t)ksel * J_;
    const float invVs = 1.f / visSum[b * K_ + ksel];

    // ---- Stage 4: OKS maps for the assigned GT ----
    for (int i = tid; i < NG_ * J_; i += 256) {
        int n = i / J_;
        int j = i - n * J_;
        float4 kv = kpS[j];
        int idx = j * NG_ + n;
        v2f xy = sA0[idx];
        float sq = sA1[idx].x;
        // folded distance: x^2+y^2 - 2x*kx - 2y*ky + kx^2+ky^2
        float d = sq + xy.x * kv.x + xy.y * kv.y + kv.z;
        float oks = (kv.w > 0.f) ? expf(-d * kv.w) : 0.f;
        out_parts[(size_t)(bc * NG_ + n) * J_ + j] = oks;
        atomicAdd(&sPerson[n], oks);
    }
    __syncthreads();
    if (tid < NG_)
        out_person[bc * NG_ + tid] = sPerson[tid] * invVs;

    // ---- Stage 5: pose_gt for the assigned GT ----
    if (tid < J_) {
        int j = tid;
        const float* kp = keypoints + (size_t)((b * K_ + ksel) * J_ + j) * 3;
        float X = kp[0], Y = kp[1], V = kp[2];
        const float* tr = transforms + b * 9;
        float gx = tr[0] * X + tr[1] * Y + tr[2];
        float gy = tr[3] * X + tr[4] * Y + tr[5];
        if (flip) gx = (float)(W_ - 1) - gx;
        float s2 = sigmas[j] * 2.f;
        float area_sel = areas[b * K_ + ksel] * s2 * s2;
        float gv = (tr[0] * tr[4]) * area_sel * ((V > 0.f) ? 1.f : 0.f);
        float* op = out_pose + (size_t)(bc * J_ + j) * 3;
        op[0] = gx; op[1] = gy; op[2] = gv;
    }
}

// ---------------------------------------------------------------------------
extern "C" void kernel_launch(void* const* d_in, const int* in_sizes, int n_in,
                              void* d_out, int out_size, void* d_ws, size_t ws_size,
                              hipStream_t stream)
{
    (void)in_sizes; (void)n_in; (void)out_size; (void)ws_size;

    const float* pose_pool      = (const float*)d_in[0];
    const float* keypoints      = (const float*)d_in[1];
    const float* areas          = (const float*)d_in[2];
    const float* transforms     = (const float*)d_in[3];
    const float* transforms_inv = (const float*)d_in[4];
    const int*   hflip          = (const int*)d_in[5];
    const float* sigmas         = (const float*)d_in[6];
    // d_in[7]=height, d_in[8]=width (compile-time constants here: 128x128)

    float* out_parts  = (float*)d_out;                       // B*C*64*J
    float* out_person = out_parts + (size_t)B_ * C_ * NG_ * J_;
    float* out_pose   = out_person + (size_t)B_ * C_ * NG_;

    float4* kp4   = (float4*)d_ws;                           // B*K*J float4
    float* visSum = (float*)((char*)d_ws + (size_t)B_ * K_ * J_ * sizeof(float4));

    oks_prep<<<(B_ * K_ + 255) / 256, 256, 0, stream>>>(
        keypoints, areas, sigmas, kp4, visSum);

    oks_main<<<B_ * C_, 256, 0, stream>>>(
        pose_pool, keypoints, areas, transforms, transforms_inv, hflip, sigmas,
        kp4, visSum, out_parts, out_person, out_pose);
}